// HMCLayer_8967891714645
// MI455X (gfx1250) — compile-verified
//
#include <hip/hip_runtime.h>
#include <hip/hip_bf16.h>
#include <math.h>

// ---------------------------------------------------------------------------
// HMC hetero-GNN layer for MI455X (gfx1250, wave32).
//  - Dense x@W GEMMs via V_WMMA_F32_16X16X4_F32 (f32 reference precision),
//    4 M-tiles per wave to amortize B-fragment loads (4 WMMA : 6 loads).
//  - Sparse attention (segment softmax + SPMM) with native f32 global atomics.
// ---------------------------------------------------------------------------

#define D128 128
#define NEG_SLOPE 0.2f

typedef float v2f __attribute__((ext_vector_type(2)));
typedef float v8f __attribute__((ext_vector_type(8)));

// ---------------------------------------------------------------------------
// GEMM: M[n,128] = X[n,128] @ W[128,128]   (row-major, f32)
// 256 threads = 8 waves; wave w owns N columns [16w, 16w+16) and computes
// FOUR 16x16 output tiles (rows blockIdx.x*64 + 16t).  The B fragment is
// shared by all 4 WMMAs per k-step.
//
// A 16x4 f32 layout : lanes 0-15 M=0..15 hold K={k,k+1}; lanes 16-31 K={k+2,k+3}
// B 4x16 f32 layout : lane%16 = N; lanes<16 rows K={k,k+1}; lanes>=16 K={k+2,k+3}
// C/D 16x16 f32     : vgpr r -> M=r (lanes 0-15) / M=r+8 (lanes 16-31), N=lane%16
//
// Out-of-range A rows are clamped to row 0 WITHOUT zeroing: a garbage A row M
// only affects output row M of that tile, which is never stored.
// ---------------------------------------------------------------------------
__global__ __launch_bounds__(256) void gemm128_wmma(
    const float* __restrict__ X, const float* __restrict__ W,
    float* __restrict__ M, int n) {
  const int wave  = threadIdx.x >> 5;
  const int lane  = threadIdx.x & 31;
  const int l15   = lane & 15;
  const int khalf = (lane >> 4) << 1;          // 0 or 2
  const int row0  = blockIdx.x * 64;
  const int ncol  = wave * 16 + l15;

  const float* xrow[4];
#pragma unroll
  for (int t = 0; t < 4; ++t) {
    int mr = row0 + t * 16 + l15;
    xrow[t] = X + (size_t)(mr < n ? mr : 0) * D128;
  }

  v8f acc[4] = {{}, {}, {}, {}};
#pragma unroll 4
  for (int k0 = 0; k0 < D128; k0 += 4) {
    v2f b;
    b.x = W[(k0 + khalf) * D128 + ncol];
    b.y = W[(k0 + khalf + 1) * D128 + ncol];
    v2f a[4];
#pragma unroll
    for (int t = 0; t < 4; ++t) {
      a[t].x = xrow[t][k0 + khalf];
      a[t].y = xrow[t][k0 + khalf + 1];
    }
#pragma unroll
    for (int t = 0; t < 4; ++t) {
      acc[t] = __builtin_amdgcn_wmma_f32_16x16x4_f32(
          false, a[t], false, b, (short)0, acc[t], false, false);
    }
  }

  const int hi8 = (lane >> 4) << 3;            // 0 or 8
#pragma unroll
  for (int t = 0; t < 4; ++t) {
    const int tr0 = row0 + t * 16;
    if (tr0 + 15 < n) {                        // whole tile in range (fast path)
#pragma unroll
      for (int r = 0; r < 8; ++r)
        M[(size_t)(tr0 + hi8 + r) * D128 + ncol] = acc[t][r];
    } else if (tr0 < n) {                      // partial tail tile
#pragma unroll
      for (int r = 0; r < 8; ++r) {
        int mr = tr0 + hi8 + r;
        if (mr < n) M[(size_t)mr * D128 + ncol] = acc[t][r];
      }
    }
  }
}

// ---------------------------------------------------------------------------
// Per-node dots: plo[i] = dot(M[i], a[0:128]); phi[i] = dot(M[i], a[128:256])
// One wave per node, float4 loads, shuffle reduction.
// ---------------------------------------------------------------------------
__global__ __launch_bounds__(256) void node_dots(
    const float* __restrict__ M, const float* __restrict__ a,
    float* __restrict__ plo, float* __restrict__ phi, int n) {
  const int wave = threadIdx.x >> 5;
  const int lane = threadIdx.x & 31;
  const int i = blockIdx.x * 8 + wave;
  if (i >= n) return;
  float4 mv = ((const float4*)(M + (size_t)i * D128))[lane];
  float4 al = ((const float4*)a)[lane];
  float4 ah = ((const float4*)(a + D128))[lane];
  float slo = mv.x * al.x + mv.y * al.y + mv.z * al.z + mv.w * al.w;
  float shi = mv.x * ah.x + mv.y * ah.y + mv.z * ah.z + mv.w * ah.w;
#pragma unroll
  for (int off = 16; off > 0; off >>= 1) {
    slo += __shfl_down(slo, off, 32);
    shi += __shfl_down(shi, off, 32);
  }
  if (lane == 0) { plo[i] = slo; phi[i] = shi; }
}

__global__ __launch_bounds__(256) void fill_neginf(float* __restrict__ p, int n) {
  int i = blockIdx.x * blockDim.x + threadIdx.x;
  if (i < n) p[i] = -__builtin_inff();
}

__device__ __forceinline__ float lrelu(float x) {
  return x > 0.0f ? x : NEG_SLOPE * x;
}

// Pass 1: segment max of lrelu(dA[iA[e]] + dB[iB[e]]) over seg[e].
__global__ __launch_bounds__(256) void edge_att_max(
    const float* __restrict__ dA, const int* __restrict__ iA,
    const float* __restrict__ dB, const int* __restrict__ iB,
    const int* __restrict__ seg, float* __restrict__ rowmax, int E) {
  int e = blockIdx.x * blockDim.x + threadIdx.x;
  if (e >= E) return;
  float v = lrelu(dA[iA[e]] + dB[iB[e]]);
  __hip_atomic_fetch_max(&rowmax[seg[e]], v, __ATOMIC_RELAXED,
                         __HIP_MEMORY_SCOPE_AGENT);
}

// Pass 2: ew[e] = exp(val - rowmax[seg]); rowsum[seg] += ew[e].
__global__ __launch_bounds__(256) void edge_exp_sum(
    const float* __restrict__ dA, const int* __restrict__ iA,
    const float* __restrict__ dB, const int* __restrict__ iB,
    const int* __restrict__ seg, const float* __restrict__ rowmax,
    float* __restrict__ rowsum, float* __restrict__ ew, int E) {
  int e = blockIdx.x * blockDim.x + threadIdx.x;
  if (e >= E) return;
  int s = seg[e];
  float v = lrelu(dA[iA[e]] + dB[iB[e]]);
  float w = __expf(v - rowmax[s]);
  ew[e] = w;
  __hip_atomic_fetch_add(&rowsum[s], w, __ATOMIC_RELAXED,
                         __HIP_MEMORY_SCOPE_AGENT);
}

// Pass 3: SPMM scatter: acc[seg[e]] += (ew/rowsum * vals) * src[gidx[e]].
// One wave per edge; each lane handles 4 contiguous dims (float4 gather).
__global__ __launch_bounds__(256) void edge_spmm(
    const float* __restrict__ ew, const float* __restrict__ rowsum,
    const float* __restrict__ vals, const int* __restrict__ seg,
    const int* __restrict__ gidx, const float* __restrict__ src,
    float* __restrict__ acc, int E) {
  const int wave = threadIdx.x >> 5;
  const int lane = threadIdx.x & 31;
  int e = blockIdx.x * 8 + wave;
  if (e >= E) return;
  int r = seg[e], c = gidx[e];
  float coef = ew[e] / rowsum[r] * vals[e];
  float4 mv = ((const float4*)(src + (size_t)c * D128))[lane];
  float* dst = acc + (size_t)r * D128 + lane * 4;
  __hip_atomic_fetch_add(dst + 0, coef * mv.x, __ATOMIC_RELAXED, __HIP_MEMORY_SCOPE_AGENT);
  __hip_atomic_fetch_add(dst + 1, coef * mv.y, __ATOMIC_RELAXED, __HIP_MEMORY_SCOPE_AGENT);
  __hip_atomic_fetch_add(dst + 2, coef * mv.z, __ATOMIC_RELAXED, __HIP_MEMORY_SCOPE_AGENT);
  __hip_atomic_fetch_add(dst + 3, coef * mv.w, __ATOMIC_RELAXED, __HIP_MEMORY_SCOPE_AGENT);
}

// P[i] += relu(acc[i])   (relu-sum algebra folds all combines into this)
__global__ __launch_bounds__(256) void accum_relu(
    float* __restrict__ P, const float* __restrict__ acc, int n) {
  int i = blockIdx.x * blockDim.x + threadIdx.x;
  if (i < n) P[i] += fmaxf(acc[i], 0.0f);
}

// ---------------------------------------------------------------------------
// Host-side orchestration
// ---------------------------------------------------------------------------
namespace {

struct Scratch {
  float *mA, *mB, *acc;
  float *dslo, *dshi, *dtlo, *dthi;
  float *rowmax, *rowsum, *ew;
};

static inline int ceil_div(int a, int b) { return (a + b - 1) / b; }

static void run_gemm(const float* X, const float* W, float* M, int n,
                     hipStream_t s) {
  gemm128_wmma<<<ceil_div(n, 64), 256, 0, s>>>(X, W, M, n);
}

static void run_dots(const float* M, const float* a, float* lo, float* hi,
                     int n, hipStream_t s) {
  node_dots<<<ceil_div(n, 8), 256, 0, s>>>(M, a, lo, hi, n);
}

// softmax + weighted SPMM + relu-accumulate into P
static void attn_agg(const float* dA, const int* iA, const float* dB,
                     const int* iB, const int* seg, const float* vals,
                     const int* gidx, const float* src, float* P, int nseg,
                     int E, const Scratch& ws, hipStream_t s) {
  fill_neginf<<<ceil_div(nseg, 256), 256, 0, s>>>(ws.rowmax, nseg);
  hipMemsetAsync(ws.rowsum, 0, (size_t)nseg * sizeof(float), s);
  edge_att_max<<<ceil_div(E, 256), 256, 0, s>>>(dA, iA, dB, iB, seg,
                                                ws.rowmax, E);
  edge_exp_sum<<<ceil_div(E, 256), 256, 0, s>>>(dA, iA, dB, iB, seg,
                                                ws.rowmax, ws.rowsum, ws.ew, E);
  hipMemsetAsync(ws.acc, 0, (size_t)nseg * D128 * sizeof(float), s);
  edge_spmm<<<ceil_div(E, 8), 256, 0, s>>>(ws.ew, ws.rowsum, vals, seg, gidx,
                                           src, ws.acc, E);
  accum_relu<<<ceil_div(nseg * D128, 256), 256, 0, s>>>(P, ws.acc, nseg * D128);
}

// HBS: m = x@W; e = lrelu(s[rows]+t[cols]); softmax over rows; spmm m[cols].
static void hbs(const float* x, const float* W, const float* a,
                const int* rows, const int* cols, const float* vals, int n,
                int E, float* P, const Scratch& ws, hipStream_t s) {
  run_gemm(x, W, ws.mA, n, s);
  run_dots(ws.mA, a, ws.dslo, ws.dshi, n, s);
  attn_agg(ws.dslo, rows, ws.dshi, cols, rows, vals, cols, ws.mA, P, n, E, ws, s);
}

// HBNS: s_msg = xs@Ws, t_msg = xt@Wt.
//  e path (target): lrelu(slo[s_cols] + thi[t_rows]), seg=t_rows, gather s_msg
//  f path (source): lrelu(tlo[t_rows] + shi[s_cols]), seg=s_cols, gather t_msg
static void hbns(const float* xs, const float* xt, const float* Ws,
                 const float* Wt, const float* a, const int* t_rows,
                 const int* s_cols, const float* vals, int n_s, int n_t, int E,
                 float* P_src, float* P_tgt, bool want_source,
                 const Scratch& ws, hipStream_t s) {
  run_gemm(xs, Ws, ws.mA, n_s, s);
  run_gemm(xt, Wt, ws.mB, n_t, s);
  run_dots(ws.mA, a, ws.dslo, ws.dshi, n_s, s);
  run_dots(ws.mB, a, ws.dtlo, ws.dthi, n_t, s);
  attn_agg(ws.dslo, s_cols, ws.dthi, t_rows, t_rows, vals, s_cols, ws.mA,
           P_tgt, n_t, E, ws, s);
  if (want_source) {
    attn_agg(ws.dtlo, t_rows, ws.dshi, s_cols, s_cols, vals, t_rows, ws.mB,
             P_src, n_s, E, ws, s);
  }
}

}  // namespace

extern "C" void kernel_launch(void* const* d_in, const int* in_sizes, int n_in,
                              void* d_out, int out_size, void* d_ws,
                              size_t ws_size, hipStream_t stream) {
  (void)n_in; (void)out_size; (void)ws_size;

  const int N0 = in_sizes[0] / D128;
  const int N1 = in_sizes[1] / D128;
  const int N2 = in_sizes[2] / D128;
  const int EA0 = in_sizes[3], EA1 = in_sizes[6], EC2 = in_sizes[9];
  const int EI1 = in_sizes[12], EI2 = in_sizes[15];
  int Nmax = N0 > N1 ? N0 : N1; if (N2 > Nmax) Nmax = N2;
  int Emax = EA0; if (EA1 > Emax) Emax = EA1; if (EC2 > Emax) Emax = EC2;
  if (EI1 > Emax) Emax = EI1; if (EI2 > Emax) Emax = EI2;

  const float* x0 = (const float*)d_in[0];
  const float* x1 = (const float*)d_in[1];
  const float* x2 = (const float*)d_in[2];
  const int *a0r = (const int*)d_in[3], *a0c = (const int*)d_in[4];
  const float* a0v = (const float*)d_in[5];
  const int *a1r = (const int*)d_in[6], *a1c = (const int*)d_in[7];
  const float* a1v = (const float*)d_in[8];
  const int *c2r = (const int*)d_in[9], *c2c = (const int*)d_in[10];
  const float* c2v = (const float*)d_in[11];
  const int *i1r = (const int*)d_in[12], *i1c = (const int*)d_in[13];
  const float* i1v = (const float*)d_in[14];
  const int *i2r = (const int*)d_in[15], *i2c = (const int*)d_in[16];
  const float* i2v = (const float*)d_in[17];

  // params flattened in dict insertion order
  const float* hbs0_l1_W = (const float*)d_in[18];
  const float* hbs0_l1_a = (const float*)d_in[19];
  const float* hbs0_l2_W = (const float*)d_in[20];
  const float* hbs0_l2_a = (const float*)d_in[21];
  const float* hbs1_l2_W = (const float*)d_in[22];
  const float* hbs1_l2_a = (const float*)d_in[23];
  const float* hbs2_l2_W = (const float*)d_in[24];
  const float* hbs2_l2_a = (const float*)d_in[25];
  const float* h01l1_Ws = (const float*)d_in[26];
  const float* h01l1_Wt = (const float*)d_in[27];
  const float* h01l1_a  = (const float*)d_in[28];
  const float* h12l1_Ws = (const float*)d_in[29];
  const float* h12l1_Wt = (const float*)d_in[30];
  const float* h12l1_a  = (const float*)d_in[31];
  const float* h01l2_Ws = (const float*)d_in[32];
  const float* h01l2_Wt = (const float*)d_in[33];
  const float* h01l2_a  = (const float*)d_in[34];
  const float* h12l2_Ws = (const float*)d_in[35];
  const float* h12l2_Wt = (const float*)d_in[36];
  const float* h12l2_a  = (const float*)d_in[37];

  // workspace layout (floats)
  float* w = (float*)d_ws;
  size_t o = 0;
  float* P0 = w + o; o += (size_t)N0 * D128;   // level-1 accumulators (= x_l1)
  float* P1 = w + o; o += (size_t)N1 * D128;
  float* P2 = w + o; o += (size_t)N2 * D128;
  Scratch ws;
  ws.mA   = w + o; o += (size_t)Nmax * D128;
  ws.mB   = w + o; o += (size_t)Nmax * D128;
  ws.acc  = w + o; o += (size_t)Nmax * D128;
  ws.dslo = w + o; o += Nmax;
  ws.dshi = w + o; o += Nmax;
  ws.dtlo = w + o; o += Nmax;
  ws.dthi = w + o; o += Nmax;
  ws.rowmax = w + o; o += Nmax;
  ws.rowsum = w + o; o += Nmax;
  ws.ew   = w + o; o += Emax;

  float* out0 = (float*)d_out;
  float* out1 = out0 + (size_t)N0 * D128;
  float* out2 = out1 + (size_t)N1 * D128;

  hipMemsetAsync(P0, 0, (size_t)N0 * D128 * sizeof(float), stream);
  hipMemsetAsync(P1, 0, (size_t)N1 * D128 * sizeof(float), stream);
  hipMemsetAsync(P2, 0, (size_t)N2 * D128 * sizeof(float), stream);
  hipMemsetAsync(d_out, 0,
                 (size_t)(N0 + N1 + N2) * D128 * sizeof(float), stream);

  // ---- level 1 ----
  hbs(x0, hbs0_l1_W, hbs0_l1_a, a0r, a0c, a0v, N0, EA0, P0, ws, stream);
  hbns(x0, x1, h01l1_Ws, h01l1_Wt, h01l1_a, i1r, i1c, i1v, N0, N1, EI1,
       /*P_src=*/P0, /*P_tgt=*/P1, /*want_source=*/true, ws, stream);
  hbns(x1, x2, h12l1_Ws, h12l1_Wt, h12l1_a, i2r, i2c, i2v, N1, N2, EI2,
       /*P_src=*/P1, /*P_tgt=*/P2, /*want_source=*/true, ws, stream);
  // P0/P1/P2 are sums of relu'd contributions => already equal to x_l1.

  // ---- level 2 ----
  hbs(P0, hbs0_l2_W, hbs0_l2_a, a0r, a0c, a0v, N0, EA0, out0, ws, stream);
  hbs(P1, hbs1_l2_W, hbs1_l2_a, a1r, a1c, a1v, N1, EA1, out1, ws, stream);
  hbs(P2, hbs2_l2_W, hbs2_l2_a, c2r, c2c, c2v, N2, EC2, out2, ws, stream);
  hbns(P0, P1, h01l2_Ws, h01l2_Wt, h01l2_a, i1r, i1c, i1v, N0, N1, EI1,
       /*P_src=*/nullptr, /*P_tgt=*/out1, /*want_source=*/false, ws, stream);
  hbns(P1, P2, h12l2_Ws, h12l2_Wt, h12l2_a, i2r, i2c, i2v, N1, N2, EI2,
       /*P_src=*/nullptr, /*P_tgt=*/out2, /*want_source=*/false, ws, stream);
  // outputs are sums of relu'd terms => final relu is the identity.
}